// variational_iDCT3D_14259291423445
// MI455X (gfx1250) — compile-verified
//
#include <hip/hip_runtime.h>
#include <math.h>

// ---------------------------------------------------------------------------
// Problem constants
// ---------------------------------------------------------------------------
#define BB   128     // batch
#define NIN  14      // predicted coefficients per axis
#define NR   6       // sampled extra frequencies per axis
#define PP   20      // NIN + NR, coefficient cube side
#define NOUT 64      // output cube side
#define PI_F 3.14159265358979323846f

// workspace layout (in floats)
#define CT_OFF   0                              // Ct table: 64*20 = 1280
#define C_OFF    2048                           // coeff cube: 128*20^3 = 1,024,000
#define ZA_OFF   (C_OFF + 1024000)              // zA: 128*400*64 = 3,276,800
#define ZB_OFF   (ZA_OFF + 3276800)             // zB: 128*20*4096 = 10,485,760

typedef __attribute__((ext_vector_type(2))) float v2f;
typedef __attribute__((ext_vector_type(8))) float v8f;

// ---------------------------------------------------------------------------
// Ct[k][n] = w(n) * cos(pi * n * (2k+1) / 128) / 128,  k<64, n<20
// ---------------------------------------------------------------------------
__global__ void k_init_ct(float* __restrict__ ct) {
    int i = blockIdx.x * blockDim.x + threadIdx.x;
    if (i < NOUT * PP) {
        int k = i / PP, n = i % PP;
        float w = (n == 0) ? 1.0f : 2.0f;
        ct[i] = w * cosf(PI_F * (float)n * (float)(2 * k + 1) / 128.0f) * (1.0f / 128.0f);
    }
}

__device__ __forceinline__ float softplus_f(float x) {
    // numerically stable log1p(exp(x))
    return (x > 0.0f) ? (x + log1pf(expf(-x))) : log1pf(expf(x));
}

// ---------------------------------------------------------------------------
// Build coefficient cube c[b][n1][n2][n3], 20^3 per batch
// ---------------------------------------------------------------------------
__global__ void k_build(const float* __restrict__ x,
                        const float* __restrict__ ploc,
                        const float* __restrict__ pscale,
                        const float* __restrict__ epsp,
                        const float* __restrict__ mu1, const float* __restrict__ rho1, const float* __restrict__ eps1,
                        const float* __restrict__ mu2, const float* __restrict__ rho2, const float* __restrict__ eps2,
                        const float* __restrict__ mu3, const float* __restrict__ rho3, const float* __restrict__ eps3,
                        float* __restrict__ c) {
    int idx = blockIdx.x * blockDim.x + threadIdx.x;
    if (idx >= BB * PP * PP * PP) return;
    int n3 = idx % PP;
    int t  = idx / PP;
    int n2 = t % PP;  t /= PP;
    int n1 = t % PP;
    int b  = t / PP;

    float val;
    if (n3 >= NIN) {                    // z3: (20,20,6)
        int i = (n1 * PP + n2) * NR + (n3 - NIN);
        val = mu3[i] + softplus_f(rho3[i]) * eps3[i];
    } else if (n2 >= NIN) {             // z2: (20,6,14)
        int i = (n1 * NR + (n2 - NIN)) * NIN + n3;
        val = mu2[i] + softplus_f(rho2[i]) * eps2[i];
    } else if (n1 >= NIN) {             // z1: (6,14,14)
        int i = ((n1 - NIN) * NIN + n2) * NIN + n3;
        val = mu1[i] + softplus_f(rho1[i]) * eps1[i];
    } else {                            // predicted block + mean-field perturbation
        int ip = (n1 * NIN + n2) * NIN + n3;
        val = x[((b * NIN + n1) * NIN + n2) * NIN + n3]
            + ploc[ip] + softplus_f(pscale[ip]) * epsp[ip];
    }
    c[idx] = val;
}

// ---------------------------------------------------------------------------
// One 16x16 fp32 WMMA tile, A = Ct (64x20), K = 20 -> 5 chained 16x16x4 ops.
// bTrans==1: B element (k,n) at Bp[n*ldb + k]; else Bp[k*ldb + n].
// ---------------------------------------------------------------------------
__device__ __forceinline__ v8f wmma_ct_k20(const float* __restrict__ T, int m0,
                                           const float* __restrict__ Bp,
                                           int ldb, int bTrans, int n0) {
    const int lane = threadIdx.x & 31;
    const int l15  = lane & 15;
    const int kh   = (lane >> 4) << 1;          // 0 (lanes 0-15) or 2 (lanes 16-31)
    v8f acc = {};
#pragma unroll
    for (int k0 = 0; k0 < PP; k0 += 4) {
        const int kk = k0 + kh;
        v2f a, b;
        a.x = T[(m0 + l15) * PP + kk];          // A(m, kk)
        a.y = T[(m0 + l15) * PP + kk + 1];      // A(m, kk+1)
        if (bTrans) {
            b.x = Bp[(size_t)(n0 + l15) * ldb + kk];
            b.y = Bp[(size_t)(n0 + l15) * ldb + kk + 1];
        } else {
            b.x = Bp[(size_t)kk * ldb + (n0 + l15)];
            b.y = Bp[(size_t)(kk + 1) * ldb + (n0 + l15)];
        }
        acc = __builtin_amdgcn_wmma_f32_16x16x4_f32(false, a, false, b,
                                                    (short)0, acc, false, false);
    }
    return acc;
}

// D(m,n) -> Dst[n*ldd + m]  (per-lane contiguous 8 floats)
__device__ __forceinline__ void store_colmajor(float* __restrict__ D, int ldd,
                                               int m0, int n0, v8f acc) {
    const int lane = threadIdx.x & 31;
    float* p = D + (size_t)(n0 + (lane & 15)) * ldd + m0 + ((lane >> 4) << 3);
#pragma unroll
    for (int v = 0; v < 8; ++v) p[v] = acc[v];
}

// D(m,n) -> Dst[m*ldd + n]  (coalesced across lanes per VGPR)
__device__ __forceinline__ void store_rowmajor(float* __restrict__ D, int ldd,
                                               int m0, int n0, v8f acc) {
    const int lane = threadIdx.x & 31;
    const int mb = m0 + ((lane >> 4) << 3);
    const int n  = n0 + (lane & 15);
#pragma unroll
    for (int v = 0; v < 8; ++v) D[(size_t)(mb + v) * ldd + n] = acc[v];
}

// ---------------------------------------------------------------------------
// Stage A: zA[b][(n1,n2)][k3] = sum_n3 Ct[k3,n3] * c[b][n1][n2][n3]
//   GEMM per b: M=64(k3), N=400(n1n2), K=20; B col-major (ldb=20)
//   output stored column-major (u*64 + k3) == zA[b][n1][n2][k3]
// ---------------------------------------------------------------------------
__global__ void k_stageA(const float* __restrict__ ct,
                         const float* __restrict__ c,
                         float* __restrict__ zA) {
    int wave = (blockIdx.x * blockDim.x + threadIdx.x) >> 5;
    int b = wave / 100;                 // 4 m-tiles * 25 n-tiles
    if (b >= BB) return;
    int t  = wave % 100;
    int mt = t / 25, nt = t % 25;
    const float* Bp = c + (size_t)b * 8000;
    v8f acc = wmma_ct_k20(ct, mt * 16, Bp, PP, 1, nt * 16);
    store_colmajor(zA + (size_t)b * 25600, NOUT, mt * 16, nt * 16, acc);
}

// ---------------------------------------------------------------------------
// Stage B: zB[b][n1][k2][k3] = sum_n2 Ct[k2,n2] * zA[b][n1][n2][k3]
//   GEMM per (b,n1): M=64(k2), N=64(k3), K=20; B row-major (ldb=64)
// ---------------------------------------------------------------------------
__global__ void k_stageB(const float* __restrict__ ct,
                         const float* __restrict__ zA,
                         float* __restrict__ zB) {
    int wave = (blockIdx.x * blockDim.x + threadIdx.x) >> 5;
    int b = wave / (PP * 16);
    if (b >= BB) return;
    int n1 = (wave / 16) % PP;
    int t  = wave & 15;
    int mt = t >> 2, nt = t & 3;
    const float* Bp = zA + (size_t)b * 25600 + (size_t)n1 * 1280;
    v8f acc = wmma_ct_k20(ct, mt * 16, Bp, NOUT, 0, nt * 16);
    store_rowmajor(zB + (size_t)b * 81920 + (size_t)n1 * 4096, NOUT,
                   mt * 16, nt * 16, acc);
}

// ---------------------------------------------------------------------------
// Stage C: out[b][k1][k2][k3] = sum_n1 Ct[k1,n1] * zB[b][n1][k2][k3]
//   GEMM per b: M=64(k1), N=4096(k2k3), K=20; B row-major (ldb=4096)
// ---------------------------------------------------------------------------
__global__ void k_stageC(const float* __restrict__ ct,
                         const float* __restrict__ zB,
                         float* __restrict__ out) {
    int wave = (blockIdx.x * blockDim.x + threadIdx.x) >> 5;
    int b = wave >> 10;                 // 4 m-tiles * 256 n-tiles = 1024 tiles/b
    if (b >= BB) return;
    int t  = wave & 1023;
    int mt = t >> 8, nt = t & 255;
    const float* Bp = zB + (size_t)b * 81920;
    v8f acc = wmma_ct_k20(ct, mt * 16, Bp, 4096, 0, nt * 16);
    store_rowmajor(out + (size_t)b * 262144, 4096, mt * 16, nt * 16, acc);
}

// ---------------------------------------------------------------------------
extern "C" void kernel_launch(void* const* d_in, const int* in_sizes, int n_in,
                              void* d_out, int out_size, void* d_ws, size_t ws_size,
                              hipStream_t stream) {
    const float* x      = (const float*)d_in[0];
    const float* ploc   = (const float*)d_in[1];
    const float* pscale = (const float*)d_in[2];
    const float* epsp   = (const float*)d_in[3];
    const float* mu1    = (const float*)d_in[4];
    const float* rho1   = (const float*)d_in[5];
    const float* eps1   = (const float*)d_in[6];
    const float* mu2    = (const float*)d_in[7];
    const float* rho2   = (const float*)d_in[8];
    const float* eps2   = (const float*)d_in[9];
    const float* mu3    = (const float*)d_in[10];
    const float* rho3   = (const float*)d_in[11];
    const float* eps3   = (const float*)d_in[12];

    float* ws  = (float*)d_ws;
    float* ct  = ws + CT_OFF;
    float* c   = ws + C_OFF;
    float* zA  = ws + ZA_OFF;
    float* zB  = ws + ZB_OFF;
    float* out = (float*)d_out;

    // Ct table: 1280 entries
    k_init_ct<<<5, 256, 0, stream>>>(ct);

    // coefficient cube: 1,024,000 entries
    k_build<<<(BB * PP * PP * PP + 255) / 256, 256, 0, stream>>>(
        x, ploc, pscale, epsp, mu1, rho1, eps1, mu2, rho2, eps2,
        mu3, rho3, eps3, c);

    // Stage A: 128 b * 100 tiles = 12800 waves / 8 waves-per-block
    k_stageA<<<1600, 256, 0, stream>>>(ct, c, zA);
    // Stage B: 128 * 20 * 16 = 40960 waves
    k_stageB<<<5120, 256, 0, stream>>>(ct, zA, zB);
    // Stage C: 128 * 1024 = 131072 waves
    k_stageC<<<16384, 256, 0, stream>>>(ct, zB, out);
}